// AutoformerBlock_22007412424958
// MI455X (gfx1250) — compile-verified
//
#include <hip/hip_runtime.h>
#include <hip/hip_bf16.h>
#include <stdint.h>

// ---------------------------------------------------------------------------
// Autoformer block for MI455X (gfx1250, wave32)
//   B=16, C=256, T=4096, POOL_K=25, TOP_K=3
//   d_out layout: [out | trend | seasonal_agg], each B*C*T fp32
// ---------------------------------------------------------------------------

#define BB 16
#define CC 256
#define TT 4096
#define NFFT 4096
#define LOG2N 12
#define POOLK 25
#define POOLH 12
#define NTHREADS 256
#define STRIP 32                      // GEMM: columns per block
#define NEG_BIG (-3.402823466e+38f)

typedef float v2f __attribute__((ext_vector_type(2)));
typedef float v8f __attribute__((ext_vector_type(8)));

// generic (flat) shared pointer -> 32-bit LDS byte offset.
// LDS flat addresses carry the LDS offset in the low 32 bits (ISA 10.2
// aperture mapping).
__device__ __forceinline__ uint32_t lds_addr32(const void* p) {
    return (uint32_t)(uintptr_t)p;
}

// ---------------------------------------------------------------------------
// Kernel 1: series decomposition (AvgPool1d k=25, pad=12, count_include_pad)
// one block per (b,c) row
// ---------------------------------------------------------------------------
__global__ __launch_bounds__(NTHREADS) void decompose_kernel(
    const float* __restrict__ x,
    float* __restrict__ trend,
    float* __restrict__ seasonal)
{
    __shared__ float s[TT];                       // 16 KB
    const int row = blockIdx.x;                   // b*C + c
    const float* xr = x + (size_t)row * TT;

    for (int i = threadIdx.x; i < TT; i += NTHREADS) s[i] = xr[i];
    __syncthreads();

    for (int i = threadIdx.x; i < TT; i += NTHREADS) {
        float acc = 0.f;
#pragma unroll
        for (int w = -POOLH; w <= POOLH; ++w) {
            int t = i + w;
            if (t >= 0 && t < TT) acc += s[t];    // zero padding outside
        }
        float tr = acc * (1.0f / (float)POOLK);
        trend[(size_t)row * TT + i]    = tr;
        seasonal[(size_t)row * TT + i] = s[i] - tr;
    }
}

// ---------------------------------------------------------------------------
// Kernel 2: FFT autocorrelation + top-3 lag shift-and-add, in place on the
// seasonal region. One block per row. Stockham radix-2 in LDS (ping-pong).
//   ac = irfft(F * conj(F)); |F|^2 is real-even, so a second *forward* FFT
//   of the power spectrum yields N*ac (scale irrelevant for top-k).
// ---------------------------------------------------------------------------
__global__ __launch_bounds__(NTHREADS) void autocorr_kernel(
    float* __restrict__ seasonal)
{
    __shared__ float lds[4 * NFFT];               // 64 KB: re0 im0 re1 im1
    __shared__ int   lags[3];
    float* re0 = lds;
    float* im0 = lds + NFFT;
    float* re1 = lds + 2 * NFFT;
    float* im1 = lds + 3 * NFFT;

    const int tid = threadIdx.x;
    float* srow = seasonal + (size_t)blockIdx.x * NFFT;

    for (int i = tid; i < NFFT; i += NTHREADS) { re0[i] = srow[i]; im0[i] = 0.f; }
    __syncthreads();

    float* srcRe = re0; float* srcIm = im0;
    float* dstRe = re1; float* dstIm = im1;

    for (int pass = 0; pass < 2; ++pass) {
        int Ns = 1;
        for (int stage = 0; stage < LOG2N; ++stage) {
#pragma unroll 2
            for (int j = tid; j < NFFT / 2; j += NTHREADS) {
                float aR = srcRe[j],            aI = srcIm[j];
                float bR = srcRe[j + NFFT / 2], bI = srcIm[j + NFFT / 2];
                int r = j & (Ns - 1);
                float ang = -6.28318530717958647692f * (float)r / (float)(2 * Ns);
                float sw, cw;
                __sincosf(ang, &sw, &cw);                  // w = e^{-i*2pi*r/(2Ns)}
                float tR = cw * bR - sw * bI;
                float tI = cw * bI + sw * bR;
                int idx = ((j - r) << 1) + r;              // (j/Ns)*2Ns + r
                dstRe[idx]      = aR + tR;  dstIm[idx]      = aI + tI;
                dstRe[idx + Ns] = aR - tR;  dstIm[idx + Ns] = aI - tI;
            }
            __syncthreads();
            float* t;
            t = srcRe; srcRe = dstRe; dstRe = t;
            t = srcIm; srcIm = dstIm; dstIm = t;
            Ns <<= 1;
        }
        if (pass == 0) {
            // power spectrum (real, even) -> input of second forward FFT
            for (int i = tid; i < NFFT; i += NTHREADS) {
                float r = srcRe[i], m = srcIm[i];
                srcRe[i] = r * r + m * m;
                srcIm[i] = 0.f;
            }
            __syncthreads();
        }
    }
    // 12 stages (even) per pass -> result sits back in buffer 0 == srcRe
    const float* ac = srcRe;                       // N * autocorr
    float* redV = dstRe;                           // free buffer -> scratch
    int*   redI = (int*)dstIm;

    float v0 = NEG_BIG, v1 = NEG_BIG, v2 = NEG_BIG;
    int   i0 = 0, i1 = 0, i2 = 0;
    for (int i = tid; i < NFFT; i += NTHREADS) {
        float v = ac[i];
        if (v > v0)      { v2 = v1; i2 = i1; v1 = v0; i1 = i0; v0 = v; i0 = i; }
        else if (v > v1) { v2 = v1; i2 = i1; v1 = v;  i1 = i; }
        else if (v > v2) { v2 = v;  i2 = i; }
    }
    redV[tid * 3 + 0] = v0; redI[tid * 3 + 0] = i0;
    redV[tid * 3 + 1] = v1; redI[tid * 3 + 1] = i1;
    redV[tid * 3 + 2] = v2; redI[tid * 3 + 2] = i2;
    __syncthreads();

    if (tid == 0) {
        float b0 = NEG_BIG, b1 = NEG_BIG, b2 = NEG_BIG;
        int   j0 = 0, j1 = 0, j2 = 0;
        for (int i = 0; i < 3 * NTHREADS; ++i) {
            float v = redV[i]; int ix = redI[i];
            if (v > b0)      { b2 = b1; j2 = j1; b1 = b0; j1 = j0; b0 = v; j0 = ix; }
            else if (v > b1) { b2 = b1; j2 = j1; b1 = v;  j1 = ix; }
            else if (v > b2) { b2 = v;  j2 = ix; }
        }
        lags[0] = j0; lags[1] = j1; lags[2] = j2;
    }
    __syncthreads();
    const int l0 = lags[0], l1 = lags[1], l2 = lags[2];

    // agg[t] = mean_k s[t - lag_k] (t >= lag_k else 0); read-then-overwrite.
    float aggv[NFFT / NTHREADS];
#pragma unroll
    for (int q = 0; q < NFFT / NTHREADS; ++q) {
        int t = tid + q * NTHREADS;
        float a = 0.f;
        if (t >= l0) a += srow[t - l0];
        if (t >= l1) a += srow[t - l1];
        if (t >= l2) a += srow[t - l2];
        aggv[q] = a * (1.0f / 3.0f);
    }
    __syncthreads();                               // all reads of srow done
#pragma unroll
    for (int q = 0; q < NFFT / NTHREADS; ++q)
        srow[tid + q * NTHREADS] = aggv[q];
}

// ---------------------------------------------------------------------------
// Kernel 3: 1x1 conv == GEMM  out_b = W(256x256) @ (trend_b + seasonal_b) + b
// Block = one batch x 32-column strip, ALL 256 output rows -> trend/seasonal
// are streamed from HBM exactly once. Strip tiles (256x32 fp32 each) are
// staged into LDS with gfx1250 async-to-LDS DMA (ASYNCcnt), with a 16B-chunk
// XOR swizzle so the two half-wave row groups of a WMMA B-operand ds_load hit
// disjoint LDS bank halves.
//   fp32 WMMA 16x16x4 layouts:
//   A: lanes 0-15 M=lane (VGPR0=K0,VGPR1=K1); lanes 16-31 M=lane-16 (K2,K3)
//   B: lanes 0-15 N=lane (K0,K1);             lanes 16-31 N=lane-16 (K2,K3)
//   C/D: VGPRv = row (v | v+8), col = lane & 15
// ---------------------------------------------------------------------------
__global__ __launch_bounds__(NTHREADS) void conv1x1_wmma_kernel(
    const float* __restrict__ Wm,       // [C_out][C_in] row-major, 256x256
    const float* __restrict__ bias,     // [256]
    const float* __restrict__ trend,    // [B][C][T]
    const float* __restrict__ seasonal, // [B][C][T]
    float* __restrict__ out)            // [B][C][T]
{
    __shared__ float ldsT[CC * STRIP];  // 32 KB trend strip   (swizzled)
    __shared__ float ldsS[CC * STRIP];  // 32 KB seasonal strip(swizzled)

    const int tid  = threadIdx.x;
    const int lane = tid & 31;
    const int wave = tid >> 5;

    const int b     = blockIdx.x >> 7;          // 16 batches
    const int strip = blockIdx.x & 127;         // 4096/32 strips
    const int n0    = strip * STRIP;

    const float* baseT = trend    + (size_t)b * CC * TT + n0;
    const float* baseS = seasonal + (size_t)b * CC * TT + n0;
    const uint32_t ldsTo = lds_addr32(ldsT);
    const uint32_t ldsSo = lds_addr32(ldsS);

    // ---- async stage: 2048 chunks of 16B per tile; 8 per thread per tile
#pragma unroll
    for (int i = 0; i < 8; ++i) {
        const int q   = tid + i * NTHREADS;     // chunk id
        const int r   = q >> 3;                 // row (K index), 0..255
        const int cq  = q & 7;                  // 16B chunk within row
        const int cqs = cq ^ (((r >> 1) & 1) * 4);   // bank-half swizzle
        const uint32_t loff = (uint32_t)((r * STRIP + cqs * 4) * 4);
        uint64_t gT = (uint64_t)(uintptr_t)(baseT + (size_t)r * TT + cq * 4);
        uint64_t gS = (uint64_t)(uintptr_t)(baseS + (size_t)r * TT + cq * 4);
        uint32_t lT = ldsTo + loff;
        uint32_t lS = ldsSo + loff;
        asm volatile("global_load_async_to_lds_b128 %0, %1, off"
                     :: "v"(lT), "v"(gT) : "memory");
        asm volatile("global_load_async_to_lds_b128 %0, %1, off"
                     :: "v"(lS), "v"(gS) : "memory");
    }
    asm volatile("s_wait_asynccnt 0x0" ::: "memory");
    __syncthreads();

    // ---- compute: 16 mt-tiles x 2 n-subtiles = 32 jobs over 8 waves
    const int lane15 = lane & 15;
    const int khalf  = (lane >> 4) << 1;        // 0 (lanes 0-15) or 2 (16-31)

    for (int job = wave; job < 32; job += 8) {
        const int mt = job >> 1;
        const int nt = job & 1;
        const int m0 = mt << 4;
        const int nc = (nt << 4) + lane15;      // column within strip (0..31)
        const int mrow = m0 + lane15;
        const v2f* Arow = (const v2f*)(Wm + mrow * CC);

        v8f acc = {};
#pragma unroll 4
        for (int k0 = 0; k0 < CC; k0 += 4) {
            const int ka = k0 + khalf;          // even
            v2f Aa = Arow[ka >> 1];             // W[mrow][ka], W[mrow][ka+1]
            const int cx = nc ^ ((( ka      >> 1) & 1) * 16);
            const int cy = nc ^ ((((ka + 1) >> 1) & 1) * 16);
            v2f Bb;
            Bb.x = ldsT[ ka      * STRIP + cx] + ldsS[ ka      * STRIP + cx];
            Bb.y = ldsT[(ka + 1) * STRIP + cy] + ldsS[(ka + 1) * STRIP + cy];
            acc = __builtin_amdgcn_wmma_f32_16x16x4_f32(
                false, Aa, false, Bb, (short)0, acc, false, false);
        }

        const int rbase = m0 + ((lane >> 4) << 3);   // m0 or m0+8
        const int ncol  = n0 + (nt << 4) + lane15;
        float* orow = out + ((size_t)b * CC + rbase) * TT + ncol;
#pragma unroll
        for (int v = 0; v < 8; ++v)
            orow[(size_t)v * TT] = acc[v] + bias[rbase + v];
    }
}

// ---------------------------------------------------------------------------
extern "C" void kernel_launch(void* const* d_in, const int* in_sizes, int n_in,
                              void* d_out, int out_size, void* d_ws, size_t ws_size,
                              hipStream_t stream) {
    (void)in_sizes; (void)n_in; (void)out_size; (void)d_ws; (void)ws_size;

    const float* x    = (const float*)d_in[0];   // [B,C,T]
    const float* Wm   = (const float*)d_in[1];   // [C,C]
    const float* bias = (const float*)d_in[2];   // [C]

    float* out = (float*)d_out;
    const size_t BCT = (size_t)BB * CC * TT;
    float* out_main     = out;
    float* out_trend    = out + BCT;
    float* out_seasonal = out + 2 * BCT;

    // 1) decompose: trend + raw seasonal straight into output regions
    decompose_kernel<<<BB * CC, NTHREADS, 0, stream>>>(x, out_trend, out_seasonal);
    // 2) FFT autocorrelation + top-3 aggregation, in place on seasonal region
    autocorr_kernel<<<BB * CC, NTHREADS, 0, stream>>>(out_seasonal);
    // 3) channel-mixing GEMM via fp32 WMMA, async-LDS staged B operand
    conv1x1_wmma_kernel<<<BB * (TT / STRIP), NTHREADS, 0, stream>>>(
        Wm, bias, out_trend, out_seasonal, out_main);
}